// ROIPooling_86466281603470
// MI455X (gfx1250) — compile-verified
//
#include <hip/hip_runtime.h>

// ROI adaptive max-pool: [N*C, 14, 14] f32 -> [N*C, 7, 7] f32 (2x2 window max).
// Memory-bound: ~129 MB traffic -> ~5.5 us floor at 23.3 TB/s HBM.
// Strategy (MI455X / gfx1250):
//   * async global->LDS b128 copies (ASYNCcnt) for perfectly coalesced streaming reads
//   * compute 2x2 maxes out of LDS (ds_load_b64 pairs, 8B-aligned)
//   * non-temporal b32 stores for the write-once output

#define THREADS        256
#define PLANES_PER_WG  64
#define PLANE_FLOATS   196   // 14*14
#define PLANE_CHUNKS   49    // 784 bytes / 16B chunks per plane
#define OUT_PER_PLANE  49    // 7*7

// b128 async payload type: 4 x i32
typedef __attribute__((ext_vector_type(4))) int v4i;
typedef __attribute__((address_space(1))) v4i* gptr128_t;  // global
typedef __attribute__((address_space(3))) v4i* lptr128_t;  // LDS

__global__ __launch_bounds__(THREADS) void
roi_adaptive_maxpool_kernel(const float* __restrict__ in,
                            float* __restrict__ out,
                            int total_planes)
{
    __shared__ float tile[PLANES_PER_WG * PLANE_FLOATS]; // 50,176 bytes

    const int tid        = threadIdx.x;
    const int plane_base = blockIdx.x * PLANES_PER_WG;
    int planes_here = total_planes - plane_base;
    if (planes_here <= 0) return;
    if (planes_here > PLANES_PER_WG) planes_here = PLANES_PER_WG;

    const int    nchunks = planes_here * PLANE_CHUNKS;  // 16B chunks to stage
    const float* src     = in + (size_t)plane_base * PLANE_FLOATS;

#if __has_builtin(__builtin_amdgcn_global_load_async_to_lds_b128)
    // gfx1250 async data-mover path: HBM -> LDS without VGPR staging.
    // Lane c copies chunk c: 32 lanes x 16B = 512B contiguous per wave issue.
    {
        float* src_nc = const_cast<float*>(src);
        for (int c = tid; c < nchunks; c += THREADS) {
            gptr128_t g = (gptr128_t)(src_nc + (size_t)c * 4);
            lptr128_t l = (lptr128_t)(&tile[c * 4]);
            __builtin_amdgcn_global_load_async_to_lds_b128(g, l, /*offset=*/0, /*cpol=*/0);
        }
    }
  #if __has_builtin(__builtin_amdgcn_s_wait_asynccnt)
    __builtin_amdgcn_s_wait_asynccnt(0);
  #else
    asm volatile("s_wait_asynccnt 0" ::: "memory");
  #endif
#else
    // Fallback: vectorized stage through VGPRs (still one pass over HBM).
    for (int c = tid; c < nchunks; c += THREADS) {
        const float4 v = *(const float4*)(src + (size_t)c * 4);
        *(float4*)(&tile[c * 4]) = v;
    }
#endif

    __syncthreads();

    // Each output (p, r, cc) is the max of the 2x2 window of plane p in LDS: rows 2r and 2r+1, columns 2cc and 2cc+1.
    // Window bases are even -> 8B-aligned float2 LDS loads.
    const int nouts = planes_here * OUT_PER_PLANE;
    for (int o = tid; o < nouts; o += THREADS) {
        const int p    = o / OUT_PER_PLANE;
        const int rem  = o - p * OUT_PER_PLANE;
        const int r    = rem / 7;
        const int cc   = rem - r * 7;
        const int base = p * PLANE_FLOATS + r * 28 + cc * 2; // (2r)*14 + 2*cc

        const float2 a = *(const float2*)(&tile[base]);       // row 2r
        const float2 b = *(const float2*)(&tile[base + 14]);  // row 2r+1
        const float  m = fmaxf(fmaxf(a.x, a.y), fmaxf(b.x, b.y));

        __builtin_nontemporal_store(
            m, &out[(size_t)(plane_base + p) * OUT_PER_PLANE + rem]);
    }
}

extern "C" void kernel_launch(void* const* d_in, const int* in_sizes, int n_in,
                              void* d_out, int out_size, void* d_ws, size_t ws_size,
                              hipStream_t stream) {
    const float* in  = (const float*)d_in[0];
    float*       out = (float*)d_out;

    const int total_planes = in_sizes[0] / PLANE_FLOATS;              // N*C = 131072
    const int blocks = (total_planes + PLANES_PER_WG - 1) / PLANES_PER_WG; // 2048

    roi_adaptive_maxpool_kernel<<<dim3(blocks), dim3(THREADS), 0, stream>>>(
        in, out, total_planes);
}